// SubgraphGNNKernel_3582002725395
// MI455X (gfx1250) — compile-verified
//
#include <hip/hip_runtime.h>
#include <math.h>

// ---------------- problem constants (match reference) ----------------
#define NN   50000
#define SS   10
#define MM   (NN * SS)      // 500000
#define EE   2000000
#define DD   80             // NIN + HOP
#define HOPD 16
#define NOUTD 128
#define LL   3

typedef __attribute__((ext_vector_type(2))) float v2f;
typedef __attribute__((ext_vector_type(8))) float v8f;

// ---------------------------------------------------------------------
// zero a float buffer (grid-stride)
__global__ void k_zero(float* __restrict__ p, long n) {
  long stride = (long)gridDim.x * blockDim.x;
  for (long i = (long)blockIdx.x * blockDim.x + threadIdx.x; i < n; i += stride)
    p[i] = 0.0f;
}

// ---------------------------------------------------------------------
// h[M,80] = concat(x[mapper], hop_table[hop+1]);  also writes hemb[M,16]
// one float4 per thread: vi in [0, M*20)
__global__ __launch_bounds__(256) void k_build_h(
    const float* __restrict__ x, const float* __restrict__ hop_table,
    const int* __restrict__ mapper, const int* __restrict__ hop,
    float* __restrict__ h, float* __restrict__ hemb) {
  long vi = (long)blockIdx.x * blockDim.x + threadIdx.x;
  if (vi >= (long)MM * 20) return;
  int row = (int)(vi / 20);
  int q   = (int)(vi % 20);
  float4 v;
  if (q < 16) {
    v = *(const float4*)(x + (long)mapper[row] * 64 + q * 4);
  } else {
    v = *(const float4*)(hop_table + (long)(hop[row] + 1) * HOPD + (q - 16) * 4);
    *(float4*)(hemb + (long)row * HOPD + (q - 16) * 4) = v;
  }
  *(float4*)(h + (long)row * DD + q * 4) = v;
}

// ---------------------------------------------------------------------
// edge message + scatter-add:  agg[dst] += h[src] + edge_attr[emap]
// one float4 chunk per thread: vi in [0, E*20)
__global__ __launch_bounds__(256) void k_edge_scatter(
    const float* __restrict__ h, const float* __restrict__ edge_attr,
    const int* __restrict__ src, const int* __restrict__ dst,
    const int* __restrict__ emap, float* __restrict__ agg) {
  long vi = (long)blockIdx.x * blockDim.x + threadIdx.x;
  if (vi >= (long)EE * 20) return;
  int e = (int)(vi / 20);
  int q = (int)(vi % 20) * 4;
  int s = src[e], d = dst[e], m = emap[e];
  float4 hv = *(const float4*)(h + (long)s * DD + q);
  float4 ev = *(const float4*)(edge_attr + (long)m * DD + q);
  float* ap = agg + (long)d * DD + q;
  unsafeAtomicAdd(ap + 0, hv.x + ev.x);
  unsafeAtomicAdd(ap + 1, hv.y + ev.y);
  unsafeAtomicAdd(ap + 2, hv.z + ev.z);
  unsafeAtomicAdd(ap + 3, hv.w + ev.w);
}

// ---------------------------------------------------------------------
// fp32 WMMA GEMM:  C[M,N] = act(A[M,K] @ W[K,N] + bias)
// block = 256 threads = 8 waves; block tile = 128 rows x N cols.
// Full W panel (K x N) staged in LDS; each wave preloads its 16-row
// A fragment into registers once (v2f af[K/4]) and sweeps all N/16
// column tiles, so A is read from HBM exactly once.
// STATS: fuse per-column sum / sum-of-squares (for batch-norm) into the
// epilogue: register accumulate -> ds_add_f32 -> 2 global atomics/col/block.
template <int K, int N, bool BIAS, bool RELU, bool STATS>
__global__ __launch_bounds__(256) void k_gemm(
    const float* __restrict__ A, const float* __restrict__ W,
    const float* __restrict__ bias, float* __restrict__ C,
    float* __restrict__ stats, int M) {
  extern __shared__ float wlds[];            // K*N floats
  __shared__ float ssum[STATS ? N : 1];
  __shared__ float ssq[STATS ? N : 1];

  const int tid      = threadIdx.x;
  const int rowBlock = blockIdx.x * 128;

  if (STATS) {
    for (int t = tid; t < N; t += 256) { ssum[t] = 0.0f; ssq[t] = 0.0f; }
  }
  for (int t = tid; t < K * N; t += 256) wlds[t] = W[t];   // [k*N + n], coalesced
  __syncthreads();

  const int wave = tid >> 5;
  const int lane = tid & 31;
  const int half = lane >> 4;     // 0: K pair {k,k+1}, 1: {k+2,k+3}
  const int lcol = lane & 15;

  int arow = rowBlock + wave * 16 + lcol;   // A-fragment row (M = lane%16)
  if (arow >= M) arow = M - 1;              // clamp; stores are guarded
  const float* Arow = A + (long)arow * K;

  // preload entire A row-fragment into registers (K/2 floats per lane)
  v2f af[K / 4];
#pragma unroll
  for (int s = 0; s < K / 4; ++s) {
    af[s].x = Arow[4 * s + 2 * half];
    af[s].y = Arow[4 * s + 2 * half + 1];
  }

#pragma unroll 1
  for (int nt = 0; nt < N; nt += 16) {
    v8f acc = {};
#pragma unroll
    for (int s = 0; s < K / 4; ++s) {
      v2f b;
      b.x = wlds[(4 * s + 2 * half) * N + nt + lcol];
      b.y = wlds[(4 * s + 2 * half + 1) * N + nt + lcol];
      acc = __builtin_amdgcn_wmma_f32_16x16x4_f32(
          false, af[s], false, b, (short)0, acc, false, false);
    }
    float bv = 0.0f;
    if (BIAS) bv = bias[nt + lcol];
    float ls = 0.0f, lq = 0.0f;
#pragma unroll
    for (int v = 0; v < 8; ++v) {
      int r = rowBlock + wave * 16 + half * 8 + v;  // D: VGPR v -> M = v + 8*half
      if (r < M) {
        float xv = acc[v] + bv;
        if (RELU) xv = fmaxf(xv, 0.0f);
        C[(long)r * N + nt + lcol] = xv;
        if (STATS) { ls += xv; lq += xv * xv; }
      }
    }
    if (STATS) {
      atomicAdd(&ssum[nt + lcol], ls);   // ds_add_f32
      atomicAdd(&ssq[nt + lcol], lq);
    }
  }

  if (STATS) {
    __syncthreads();
    for (int t = tid; t < N; t += 256) {
      unsafeAtomicAdd(&stats[t], ssum[t]);
      unsafeAtomicAdd(&stats[N + t], ssq[t]);
    }
  }
}

// ---------------------------------------------------------------------
// h = (h2 - mu) * rsqrt(var + eps) * g + b + h   (batch stats, residual)
// also zeroes the next layer's agg buffer (same index space) if given.
__global__ __launch_bounds__(256) void k_bn_residual(
    const float* __restrict__ h2, const float* __restrict__ stats,
    const float* __restrict__ g, const float* __restrict__ b,
    float* __restrict__ h, float* __restrict__ aggzero) {
  long vi = (long)blockIdx.x * blockDim.x + threadIdx.x;
  if (vi >= (long)MM * 20) return;
  int c = (int)(vi % 20) * 4;
  const float invM = 1.0f / (float)MM;
  float4 v  = *(const float4*)(h2 + vi * 4);
  float4 ho = *(const float4*)(h + vi * 4);
  float r[4] = {v.x, v.y, v.z, v.w};
  float o[4] = {ho.x, ho.y, ho.z, ho.w};
#pragma unroll
  for (int j = 0; j < 4; ++j) {
    float mu  = stats[c + j] * invM;
    float var = stats[DD + c + j] * invM - mu * mu;
    r[j] = (r[j] - mu) * rsqrtf(var + 1e-5f) * g[c + j] + b[c + j] + o[j];
  }
  float4 w4 = {r[0], r[1], r[2], r[3]};
  *(float4*)(h + vi * 4) = w4;
  if (aggzero) {
    float4 z4 = {0.0f, 0.0f, 0.0f, 0.0f};
    *(float4*)(aggzero + vi * 4) = z4;
  }
}

// ---------------------------------------------------------------------
// centroid + subgraph terms: out[sub_batch[i]] += mask*hoe*gc + st*gs
// gates computed on the fly from hemb (K=16 dot).
__global__ __launch_bounds__(256) void k_scatter_cs(
    const float* __restrict__ hoe, const float* __restrict__ st,
    const float* __restrict__ hemb,
    const float* __restrict__ gcW, const float* __restrict__ gcB,
    const float* __restrict__ gsW, const float* __restrict__ gsB,
    const int* __restrict__ mapper, const int* __restrict__ sub_batch,
    float* __restrict__ out) {
  long idx = (long)blockIdx.x * blockDim.x + threadIdx.x;
  if (idx >= (long)MM * NOUTD) return;
  int i = (int)(idx >> 7);
  int c = (int)(idx & 127);
  const float* he = hemb + (long)i * HOPD;
  float dc = gcB[c], ds = gsB[c];
#pragma unroll
  for (int k = 0; k < HOPD; ++k) {
    float hv = he[k];
    dc = fmaf(hv, gcW[k * NOUTD + c], dc);
    ds = fmaf(hv, gsW[k * NOUTD + c], ds);
  }
  float gc = 1.0f / (1.0f + __expf(-dc));
  float gs = 1.0f / (1.0f + __expf(-ds));
  float val = st[idx] * gs;
  if (mapper[i] == sub_batch[i]) val += hoe[idx] * gc;
  unsafeAtomicAdd(out + (long)sub_batch[i] * NOUTD + c, val);
}

// context term: out[mapper[i]] += ct*gx
__global__ __launch_bounds__(256) void k_scatter_ct(
    const float* __restrict__ ct, const float* __restrict__ hemb,
    const float* __restrict__ gxW, const float* __restrict__ gxB,
    const int* __restrict__ mapper, float* __restrict__ out) {
  long idx = (long)blockIdx.x * blockDim.x + threadIdx.x;
  if (idx >= (long)MM * NOUTD) return;
  int i = (int)(idx >> 7);
  int c = (int)(idx & 127);
  const float* he = hemb + (long)i * HOPD;
  float dx = gxB[c];
#pragma unroll
  for (int k = 0; k < HOPD; ++k)
    dx = fmaf(he[k], gxW[k * NOUTD + c], dx);
  float gx = 1.0f / (1.0f + __expf(-dx));
  unsafeAtomicAdd(out + (long)mapper[i] * NOUTD + c, ct[idx] * gx);
}

// ---------------------------------------------------------------------
extern "C" void kernel_launch(void* const* d_in, const int* in_sizes, int n_in,
                              void* d_out, int out_size, void* d_ws, size_t ws_size,
                              hipStream_t stream) {
  const float* x         = (const float*)d_in[0];
  const float* edge_attr = (const float*)d_in[1];
  const float* hop_table = (const float*)d_in[2];
  const float* conv_W    = (const float*)d_in[3];   // [3,80,80]
  const float* bn_g      = (const float*)d_in[4];
  const float* bn_b      = (const float*)d_in[5];
  const float* oeW0 = (const float*)d_in[6];  const float* oeb0 = (const float*)d_in[7];
  const float* oeW1 = (const float*)d_in[8];  const float* oeb1 = (const float*)d_in[9];
  const float* oeW2 = (const float*)d_in[10]; const float* oeb2 = (const float*)d_in[11];
  const float* stW0 = (const float*)d_in[12]; const float* stb0 = (const float*)d_in[13];
  const float* stW1 = (const float*)d_in[14]; const float* stb1 = (const float*)d_in[15];
  const float* stW2 = (const float*)d_in[16]; const float* stb2 = (const float*)d_in[17];
  const float* ctW0 = (const float*)d_in[18]; const float* ctb0 = (const float*)d_in[19];
  const float* ctW1 = (const float*)d_in[20]; const float* ctb1 = (const float*)d_in[21];
  const float* ctW2 = (const float*)d_in[22]; const float* ctb2 = (const float*)d_in[23];
  const float* gcW  = (const float*)d_in[24]; const float* gcB  = (const float*)d_in[25];
  const float* gsW  = (const float*)d_in[26]; const float* gsB  = (const float*)d_in[27];
  const float* gxW  = (const float*)d_in[28]; const float* gxB  = (const float*)d_in[29];
  const int* nodes_mapper = (const int*)d_in[30];
  const int* edge_index   = (const int*)d_in[31];   // [2,E]
  const int* edges_mapper = (const int*)d_in[32];
  const int* sub_batch    = (const int*)d_in[33];
  const int* hop_ind      = (const int*)d_in[34];
  float* out = (float*)d_out;

  // workspace layout (floats): h | bufA | bufB | bufC | hemb | stats
  float* ws    = (float*)d_ws;
  float* h     = ws;
  float* bufA  = h    + (size_t)MM * DD;
  float* bufB  = bufA + (size_t)MM * NOUTD;
  float* bufC  = bufB + (size_t)MM * NOUTD;
  float* hemb  = bufC + (size_t)MM * NOUTD;
  float* stats = hemb + (size_t)MM * HOPD;

  const int* src = edge_index;
  const int* dst = edge_index + EE;

  const unsigned gemmBlocks = (MM + 127) / 128;   // 3907
  const size_t lds80x80   = (size_t)DD * DD * sizeof(float);
  const size_t lds80x128  = (size_t)DD * NOUTD * sizeof(float);
  const size_t lds128x128 = (size_t)NOUTD * NOUTD * sizeof(float);

  // 1) build h / hemb
  {
    long n = (long)MM * 20;
    k_build_h<<<(unsigned)((n + 255) / 256), 256, 0, stream>>>(
        x, hop_table, nodes_mapper, hop_ind, h, hemb);
  }

  // 2) GNN layers
  k_zero<<<4096, 256, 0, stream>>>(bufA, (long)MM * DD);
  for (int l = 0; l < LL; ++l) {
    k_zero<<<1, 256, 0, stream>>>(stats, 2 * DD);
    {
      long n = (long)EE * 20;
      k_edge_scatter<<<(unsigned)((n + 255) / 256), 256, 0, stream>>>(
          h, edge_attr, src, dst, edges_mapper, bufA);
    }
    // h2 = relu(agg @ conv_W[l]), with fused batch-stat accumulation
    k_gemm<DD, DD, false, true, true><<<gemmBlocks, 256, lds80x80, stream>>>(
        bufA, conv_W + (size_t)l * DD * DD, nullptr, bufB, stats, MM);
    {
      long n = (long)MM * 20;
      float* az = (l + 1 < LL) ? bufA : nullptr;   // pre-zero next layer's agg
      k_bn_residual<<<(unsigned)((n + 255) / 256), 256, 0, stream>>>(
          bufB, stats, bn_g + l * DD, bn_b + l * DD, h, az);
    }
  }

  // 3) output encoder MLP: h[M,80] -> bufA (h_oe)
  k_gemm<DD,    NOUTD, true, true,  false><<<gemmBlocks, 256, lds80x128,  stream>>>(h,    oeW0, oeb0, bufA, nullptr, MM);
  k_gemm<NOUTD, NOUTD, true, true,  false><<<gemmBlocks, 256, lds128x128, stream>>>(bufA, oeW1, oeb1, bufB, nullptr, MM);
  k_gemm<NOUTD, NOUTD, true, false, false><<<gemmBlocks, 256, lds128x128, stream>>>(bufB, oeW2, oeb2, bufA, nullptr, MM);

  // 4) subgraph-token MLP -> bufB
  k_gemm<NOUTD, NOUTD, true, true, false><<<gemmBlocks, 256, lds128x128, stream>>>(bufA, stW0, stb0, bufB, nullptr, MM);
  k_gemm<NOUTD, NOUTD, true, true, false><<<gemmBlocks, 256, lds128x128, stream>>>(bufB, stW1, stb1, bufC, nullptr, MM);
  k_gemm<NOUTD, NOUTD, true, true, false><<<gemmBlocks, 256, lds128x128, stream>>>(bufC, stW2, stb2, bufB, nullptr, MM);

  // 5) zero output, scatter centroid + subgraph terms
  k_zero<<<4096, 256, 0, stream>>>(out, (long)NN * NOUTD);
  {
    long n = (long)MM * NOUTD;
    k_scatter_cs<<<(unsigned)((n + 255) / 256), 256, 0, stream>>>(
        bufA, bufB, hemb, gcW, gcB, gsW, gsB, nodes_mapper, sub_batch, out);
  }

  // 6) context MLP -> bufB, scatter context term
  k_gemm<NOUTD, NOUTD, true, true, false><<<gemmBlocks, 256, lds128x128, stream>>>(bufA, ctW0, ctb0, bufB, nullptr, MM);
  k_gemm<NOUTD, NOUTD, true, true, false><<<gemmBlocks, 256, lds128x128, stream>>>(bufB, ctW1, ctb1, bufC, nullptr, MM);
  k_gemm<NOUTD, NOUTD, true, true, false><<<gemmBlocks, 256, lds128x128, stream>>>(bufC, ctW2, ctb2, bufB, nullptr, MM);
  {
    long n = (long)MM * NOUTD;
    k_scatter_ct<<<(unsigned)((n + 255) / 256), 256, 0, stream>>>(
        bufB, hemb, gxW, gxB, nodes_mapper, out);
  }
}